// Colar_static_63934883168892
// MI455X (gfx1250) — compile-verified
//
#include <hip/hip_runtime.h>
#include <hip/hip_bf16.h>
#include <math.h>

// ---------------- problem constants ----------------
#define B_    2048
#define T_    8
#define CIN_  2048
#define CH_   1024
#define MM_   32
#define NCLS_ 21
#define NM_   (NCLS_ * MM_)   // 672
#define EPSV  1e-8f

typedef float v2f __attribute__((ext_vector_type(2)));
typedef float v8f __attribute__((ext_vector_type(8)));

// ---------------- CDNA5 async global->LDS helpers (ASYNCcnt-tracked DMA path) ----------------
__device__ __forceinline__ void async_ld_b128(unsigned lds_byte_addr, const float* g) {
    // GLOBAL_LOAD_ASYNC_TO_LDS_B128: per-lane 16B, memory -> LDS, no VGPR staging
    asm volatile("global_load_async_to_lds_b128 %0, %1, off"
                 :: "v"(lds_byte_addr), "v"(g)
                 : "memory");
}
__device__ __forceinline__ void wait_async0() {
    asm volatile("s_wait_asynccnt 0x0" ::: "memory");
}

// ---------------- generic fp32 WMMA GEMM: C[M x N] = A[M x K] @ W[N x K]^T + bias ----------------
// BM=128, BN=32, BK=16. Block = 256 threads = 8 waves, waves 4(M) x 2(N),
// each wave owns a 32x16 tile = 2 x v_wmma_f32_16x16x4_f32 accumulators.
// Tiles stream via double-buffered GLOBAL_LOAD_ASYNC_TO_LDS_B128.
#define BM 128
#define BN 32
#define BK 16
#define LDSS 20   // padded LDS row stride: 80B rows -> B128-aligned stores, even (v2f reads), 16 distinct banks

__global__ __launch_bounds__(256) void gemm_f32_wmma(
    const float* __restrict__ A, int lda,
    const float* __restrict__ W, int ldw,
    float* __restrict__ C, int ldc,
    int K,
    const float* __restrict__ bias_row,   // indexed by output row (or null)
    const float* __restrict__ bias_col)   // indexed by output col (or null)
{
    __shared__ float As[2 * BM * LDSS];
    __shared__ float Ws[2 * BN * LDSS];

    const int tid  = threadIdx.x;
    const int blkM = blockIdx.x * BM;
    const int blkN = blockIdx.y * BN;
    const int wid  = tid >> 5;
    const int lane = tid & 31;
    const int l16  = lane & 15;
    const int h    = lane >> 4;           // lane half selects K-pair
    const int wm   = (wid >> 1) * 32;     // 0,32,64,96
    const int wn   = (wid & 1) * 16;      // 0,16

    v8f c0 = {0.f,0.f,0.f,0.f,0.f,0.f,0.f,0.f};
    v8f c1 = {0.f,0.f,0.f,0.f,0.f,0.f,0.f,0.f};

    // tile-load mapping: one B128 (4 floats of K) per issue
    const int ar0 = tid >> 2;             // rows 0..63
    const int ar1 = ar0 + 64;             // rows 64..127
    const int ac  = (tid & 3) * 4;        // k sub-offset 0,4,8,12
    const float* Aptr0 = A + (size_t)(blkM + ar0) * lda + ac;
    const float* Aptr1 = A + (size_t)(blkM + ar1) * lda + ac;
    const float* Wptr  = W + (size_t)(blkN + (tid >> 2)) * ldw + ac; // valid for tid<128

    const unsigned As_addr = (unsigned)(size_t)(&As[0]);
    const unsigned Ws_addr = (unsigned)(size_t)(&Ws[0]);
    const unsigned a0_off = (unsigned)((ar0 * LDSS + ac) * 4);
    const unsigned a1_off = (unsigned)((ar1 * LDSS + ac) * 4);
    const unsigned w_off  = (unsigned)(((tid >> 2) * LDSS + ac) * 4);
    const unsigned As_buf = (unsigned)(BM * LDSS * 4);
    const unsigned Ws_buf = (unsigned)(BN * LDSS * 4);

    // prologue: fill buffer 0
    async_ld_b128(As_addr + a0_off, Aptr0);
    async_ld_b128(As_addr + a1_off, Aptr1);
    if (tid < 128) async_ld_b128(Ws_addr + w_off, Wptr);

    int cur = 0;
    for (int k0 = 0; k0 < K; k0 += BK) {
        wait_async0();       // this wave's DMA into buf[cur] complete
        __syncthreads();     // all waves' fills visible; buf[cur^1] free to overwrite

        if (k0 + BK < K) {   // kick off next chunk while computing this one
            const unsigned ab = As_addr + (cur ? 0u : As_buf);
            const unsigned wb = Ws_addr + (cur ? 0u : Ws_buf);
            async_ld_b128(ab + a0_off, Aptr0 + k0 + BK);
            async_ld_b128(ab + a1_off, Aptr1 + k0 + BK);
            if (tid < 128) async_ld_b128(wb + w_off, Wptr + k0 + BK);
        }

        const float* Asc = As + cur * (BM * LDSS);
        const float* Wsc = Ws + cur * (BN * LDSS);
#pragma unroll
        for (int kk = 0; kk < 4; ++kk) {
            const int koff = kk * 4 + 2 * h;    // even -> v2f aligned
            v2f bf  = *reinterpret_cast<const v2f*>(&Wsc[(wn + l16) * LDSS + koff]);
            v2f af0 = *reinterpret_cast<const v2f*>(&Asc[(wm      + l16) * LDSS + koff]);
            v2f af1 = *reinterpret_cast<const v2f*>(&Asc[(wm + 16 + l16) * LDSS + koff]);
            c0 = __builtin_amdgcn_wmma_f32_16x16x4_f32(false, af0, false, bf, (short)0, c0, false, false);
            c1 = __builtin_amdgcn_wmma_f32_16x16x4_f32(false, af1, false, bf, (short)0, c1, false, false);
        }
        cur ^= 1;
    }

    const int col = blkN + wn + l16;
    const float bcol = bias_col ? bias_col[col] : 0.0f;
#pragma unroll
    for (int r = 0; r < 8; ++r) {
        const int m0 = blkM + wm + h * 8 + r;   // C layout: VGPR r holds M = r + 8*half
        const int m1 = m0 + 16;
        const float br0 = bias_row ? bias_row[m0] : 0.0f;
        const float br1 = bias_row ? bias_row[m1] : 0.0f;
        C[(size_t)m0 * ldc + col] = c0[r] + bcol + br0;
        C[(size_t)m1 * ldc + col] = c1[r] + bcol + br1;
    }
}

// ---------------- pack static_feat (n,c,m) -> SFt[(n*32+m), c] ----------------
__global__ __launch_bounds__(256) void pack_sft_kernel(const float* __restrict__ sf,
                                                       float* __restrict__ sft)
{
    const int idx = blockIdx.x * 256 + threadIdx.x;      // < 672*2048
    const int nm = idx >> 11;
    const int c  = idx & 2047;
    const int n  = nm >> 5;
    const int m  = nm & 31;
    sft[idx] = sf[(size_t)n * (CIN_ * MM_) + (size_t)c * MM_ + m];
}

// ---------------- row-normalize k -> kn (in place), rows of length CH ----------------
__global__ __launch_bounds__(256) void rownorm_kernel(float* __restrict__ k)
{
    const int b = blockIdx.x;
    const int tid = threadIdx.x;
    __shared__ float red[256];
    float* row = k + (size_t)b * CH_;
    float s = 0.f;
    for (int o = tid; o < CH_; o += 256) { float x = row[o]; s += x * x; }
    red[tid] = s; __syncthreads();
    for (int st = 128; st > 0; st >>= 1) { if (tid < st) red[tid] += red[tid + st]; __syncthreads(); }
    const float inv = 1.0f / fmaxf(sqrtf(red[0]), EPSV);
    for (int o = tid; o < CH_; o += 256) row[o] *= inv;
}

// ---------------- per-column stats over EkM/EvM (1024 x 672) ----------------
// norms[j] = ||EkM[:,j]||,  u[j] = sum_o Ww[o]*EvM[o,j]
__global__ __launch_bounds__(256) void colstats_kernel(const float* __restrict__ EkM,
                                                       const float* __restrict__ EvM,
                                                       const float* __restrict__ Ww,
                                                       float* __restrict__ norms,
                                                       float* __restrict__ u)
{
    const int j = blockIdx.x;
    const int tid = threadIdx.x;
    __shared__ float r1[256], r2[256];
    float s2 = 0.f, su = 0.f;
    for (int o = tid; o < CH_; o += 256) {
        const float ek = EkM[(size_t)o * NM_ + j];
        s2 += ek * ek;
        su += Ww[o] * EvM[(size_t)o * NM_ + j];
    }
    r1[tid] = s2; r2[tid] = su; __syncthreads();
    for (int st = 128; st > 0; st >>= 1) {
        if (tid < st) { r1[tid] += r1[tid + st]; r2[tid] += r2[tid + st]; }
        __syncthreads();
    }
    if (tid == 0) { norms[j] = sqrtf(r1[0]); u[j] = r2[0]; }
}

// ---------------- EknT[j,o] = EkM[o,j] / max(norms[j], eps) ----------------
__global__ __launch_bounds__(256) void transpose_norm_kernel(const float* __restrict__ EkM,
                                                             const float* __restrict__ norms,
                                                             float* __restrict__ EknT)
{
    const int idx = blockIdx.x * 256 + threadIdx.x;  // < 672*1024
    const int j = idx >> 10;
    const int o = idx & 1023;
    EknT[idx] = EkM[(size_t)o * NM_ + j] / fmaxf(norms[j], EPSV);
}

// ---------------- per-row softmax chain: cos(B x 672) -> q (in place) ----------------
// One wave per row b; lane owns m index. w = softmax_m(cos); t[n] = u[n,:].w + bw;
// fw = softmax_n(t); q = fw[n]*w[n,m].
__global__ __launch_bounds__(32) void softmax_q_kernel(float* __restrict__ cosq,
                                                       const float* __restrict__ u,
                                                       const float* __restrict__ bw)
{
    const int b = blockIdx.x;
    const int lane = threadIdx.x;
    float* row = cosq + (size_t)b * NM_;
    float wreg[NCLS_], t[NCLS_];
#pragma unroll
    for (int n = 0; n < NCLS_; ++n) {
        float v = row[n * MM_ + lane];
        float mx = v;
        for (int o = 16; o > 0; o >>= 1) mx = fmaxf(mx, __shfl_xor(mx, o, 32));
        float e = expf(v - mx);
        float s = e;
        for (int o = 16; o > 0; o >>= 1) s += __shfl_xor(s, o, 32);
        const float w = e / s;
        wreg[n] = w;
        float tp = u[n * MM_ + lane] * w;
        for (int o = 16; o > 0; o >>= 1) tp += __shfl_xor(tp, o, 32);
        t[n] = tp;
    }
    const float bwv = bw[0];
    float mx = -3.4e38f;
#pragma unroll
    for (int n = 0; n < NCLS_; ++n) mx = fmaxf(mx, t[n] + bwv);
    float s = 0.f; float f[NCLS_];
#pragma unroll
    for (int n = 0; n < NCLS_; ++n) { f[n] = expf(t[n] + bwv - mx); s += f[n]; }
    const float inv = 1.0f / s;
#pragma unroll
    for (int n = 0; n < NCLS_; ++n) row[n * MM_ + lane] = (f[n] * inv) * wreg[n];
}

// ---------------- final: out[b,j] = bout[j] + sum_c relu(h[b,c]) * Wout[j,c] ----------------
// h = concat(v, fE); one block per b.
__global__ __launch_bounds__(256) void final_out_kernel(const float* __restrict__ v,
                                                        const float* __restrict__ fE,
                                                        const float* __restrict__ Wout,
                                                        const float* __restrict__ bout,
                                                        float* __restrict__ out)
{
    const int b = blockIdx.x;
    const int tid = threadIdx.x;
    __shared__ float red[256];
    float acc[NCLS_];
#pragma unroll
    for (int j = 0; j < NCLS_; ++j) acc[j] = 0.f;

    for (int c = tid; c < 2 * CH_; c += 256) {
        float hv = (c < CH_) ? v[(size_t)b * CH_ + c] : fE[(size_t)b * CH_ + (c - CH_)];
        hv = fmaxf(hv, 0.0f);
#pragma unroll
        for (int j = 0; j < NCLS_; ++j) acc[j] += hv * Wout[(size_t)j * (2 * CH_) + c];
    }
#pragma unroll
    for (int j = 0; j < NCLS_; ++j) {
        red[tid] = acc[j]; __syncthreads();
        for (int st = 128; st > 0; st >>= 1) { if (tid < st) red[tid] += red[tid + st]; __syncthreads(); }
        if (tid == 0) out[(size_t)b * NCLS_ + j] = red[0] + bout[j];
        __syncthreads();
    }
}

// ---------------- host orchestration ----------------
extern "C" void kernel_launch(void* const* d_in, const int* in_sizes, int n_in,
                              void* d_out, int out_size, void* d_ws, size_t ws_size,
                              hipStream_t stream) {
    const float* x    = (const float*)d_in[0];
    const float* sf   = (const float*)d_in[1];
    const float* Wk   = (const float*)d_in[2];
    const float* bk   = (const float*)d_in[3];
    const float* Wv   = (const float*)d_in[4];
    const float* bv   = (const float*)d_in[5];
    const float* WEk  = (const float*)d_in[6];
    const float* bEk  = (const float*)d_in[7];
    const float* WEv  = (const float*)d_in[8];
    const float* bEv  = (const float*)d_in[9];
    const float* Ww   = (const float*)d_in[10];
    const float* bw   = (const float*)d_in[11];
    const float* Wout = (const float*)d_in[12];
    const float* bout = (const float*)d_in[13];
    float* out = (float*)d_out;
    float* ws  = (float*)d_ws;

    // workspace layout (floats, 256B-aligned offsets)
    const size_t off_k    = 0;                     // 2048*1024 (k, then kn in place)
    const size_t off_v    = off_k    + 2097152;    // 2048*1024
    const size_t off_sft  = off_v    + 2097152;    // 672*2048 (later reused as EknT 672*1024)
    const size_t off_ekm  = off_sft  + 1376256;    // 1024*672
    const size_t off_evm  = off_ekm  + 688128;     // 1024*672
    const size_t off_norm = off_evm  + 688128;     // 672 (pad 704)
    const size_t off_u    = off_norm + 704;        // 672 (pad 704)
    const size_t off_cos  = off_u    + 704;        // 2048*672 (q in place)
    const size_t off_fE   = off_cos  + 1376256;    // 2048*1024

    float* k_buf  = ws + off_k;
    float* v_buf  = ws + off_v;
    float* sft    = ws + off_sft;   // SFt, then EknT
    float* ekm    = ws + off_ekm;
    float* evm    = ws + off_evm;
    float* norms  = ws + off_norm;
    float* u_buf  = ws + off_u;
    float* cosq   = ws + off_cos;   // cos, then q
    float* fE     = ws + off_fE;

    const float* xl = x + (size_t)(T_ - 1) * CIN_;  // last timestep, row stride T_*CIN_

    // k = xl @ Wk^T + bk ; v = xl @ Wv^T + bv
    gemm_f32_wmma<<<dim3(B_ / BM, CH_ / BN), 256, 0, stream>>>(
        xl, T_ * CIN_, Wk, CIN_, k_buf, CH_, CIN_, nullptr, bk);
    gemm_f32_wmma<<<dim3(B_ / BM, CH_ / BN), 256, 0, stream>>>(
        xl, T_ * CIN_, Wv, CIN_, v_buf, CH_, CIN_, nullptr, bv);

    // kn in place
    rownorm_kernel<<<B_, 256, 0, stream>>>(k_buf);

    // pack static_feat -> SFt (672 x 2048)
    pack_sft_kernel<<<(NM_ * CIN_) / 256, 256, 0, stream>>>(sf, sft);

    // EkM(1024x672) = WEk @ SFt^T + bEk(row) ; EvM likewise
    gemm_f32_wmma<<<dim3(CH_ / BM, NM_ / BN), 256, 0, stream>>>(
        WEk, CIN_, sft, CIN_, ekm, NM_, CIN_, bEk, nullptr);
    gemm_f32_wmma<<<dim3(CH_ / BM, NM_ / BN), 256, 0, stream>>>(
        WEv, CIN_, sft, CIN_, evm, NM_, CIN_, bEv, nullptr);

    // column norms of EkM and u = Ww @ EvM
    colstats_kernel<<<NM_, 256, 0, stream>>>(ekm, evm, Ww, norms, u_buf);

    // EknT (672 x 1024), reusing SFt buffer
    transpose_norm_kernel<<<(NM_ * CH_) / 256, 256, 0, stream>>>(ekm, norms, sft);

    // cos (2048 x 672) = kn @ EknT^T
    gemm_f32_wmma<<<dim3(B_ / BM, NM_ / BN), 256, 0, stream>>>(
        k_buf, CH_, sft, CH_, cosq, NM_, CH_, nullptr, nullptr);

    // softmax chain: cos -> q (in place)
    softmax_q_kernel<<<B_, 32, 0, stream>>>(cosq, u_buf, bw);

    // fE (2048 x 1024) = q @ EvM^T
    gemm_f32_wmma<<<dim3(B_ / BM, CH_ / BN), 256, 0, stream>>>(
        cosq, NM_, evm, NM_, fE, CH_, NM_, nullptr, nullptr);

    // out = relu([v, fE]) @ Wout^T + bout
    final_out_kernel<<<B_, 256, 0, stream>>>(v_buf, fE, Wout, bout, out);
}